// MeanSAGE_81836306858015
// MI455X (gfx1250) — compile-verified
//
#include <hip/hip_runtime.h>
#include <hip/hip_bf16.h>

#define N_NODES 40000
#define N_EDGES 640000
#define IN_F 128
#define H_F 128
#define N_CLS 16
#define KDIM 256   // 2*128 concat dimension for every layer

typedef __attribute__((ext_vector_type(2))) float v2f;
typedef __attribute__((ext_vector_type(8))) float v8f;

// Guaranteed native fp32 atomic add (no CAS loop), device scope, no return.
// Tracked by STOREcnt; s_endpgm's implicit wait-idle covers completion.
__device__ __forceinline__ void gatomic_add_f32(float* p, float v) {
    asm volatile("global_atomic_add_f32 %0, %1, off scope:SCOPE_DEV"
                 :: "v"(p), "v"(v) : "memory");
}

// ---------------------------------------------------------------- zero fill
__global__ void sage_zero_kernel(float* __restrict__ p, int n) {
    int i = blockIdx.x * blockDim.x + threadIdx.x;
    int stride = gridDim.x * blockDim.x;
    for (; i < n; i += stride) p[i] = 0.0f;
}

// ------------------------------------------------- edge-parallel aggregation
// one wave32 per edge: lane L handles features [4L, 4L+4)
__global__ void sage_aggregate_kernel(const float* __restrict__ h,
                                      const float* __restrict__ w,
                                      const int* __restrict__ src,
                                      const int* __restrict__ dst,
                                      float* __restrict__ s,
                                      float* __restrict__ deg) {
    int e    = (blockIdx.x * blockDim.x + threadIdx.x) >> 5;
    int lane = threadIdx.x & 31;
    if (e >= N_EDGES) return;                 // wave-uniform
    int   sid = src[e];
    int   did = dst[e];
    float ww  = w[e];
    float4 v = ((const float4*)(h + (size_t)sid * IN_F))[lane];
    float* sp = s + (size_t)did * IN_F + lane * 4;
    gatomic_add_f32(sp + 0, v.x * ww);
    gatomic_add_f32(sp + 1, v.y * ww);
    gatomic_add_f32(sp + 2, v.z * ww);
    gatomic_add_f32(sp + 3, v.w * ww);
    if (lane == 0) gatomic_add_f32(deg + did, 1.0f);
}

// ---------------------------------------------------------- WMMA core (fp32)
// computes one 16x16 tile: rows mTile*16.., cols nTile*16.. of
//   concat(hc, s/deg)[*,256] @ W[ncols,256]^T
// lane layout per CDNA5 ISA (16x16x4 f32):
//   A: lane l -> row l&15, k-pair k0+2*(l>>4);  B: lane l -> W row (=col) l&15
__device__ __forceinline__ v8f wmma_tile_256(const float* __restrict__ hc,
                                             const float* __restrict__ s,
                                             float invd, int arow, int brow,
                                             const float* __restrict__ W,
                                             int half) {
    const float* pA = hc + (size_t)arow * IN_F + 2 * half;   // k in [0,128)
    const float* pS = s  + (size_t)arow * IN_F + 2 * half;   // k in [128,256)
    const float* pB = W  + (size_t)brow * KDIM + 2 * half;   // contiguous full K
    v8f c = {0.f, 0.f, 0.f, 0.f, 0.f, 0.f, 0.f, 0.f};
    #pragma unroll 8
    for (int k0 = 0; k0 < IN_F; k0 += 4) {                   // self features h
        v2f a = *(const v2f*)pA;
        v2f b = *(const v2f*)pB;
        c = __builtin_amdgcn_wmma_f32_16x16x4_f32(false, a, false, b,
                                                  (short)0, c, false, false);
        pA += 4; pB += 4;
    }
    #pragma unroll 8
    for (int k0 = 0; k0 < IN_F; k0 += 4) {                   // neighbor mean
        v2f a = *(const v2f*)pS;
        a = a * invd;
        v2f b = *(const v2f*)pB;
        c = __builtin_amdgcn_wmma_f32_16x16x4_f32(false, a, false, b,
                                                  (short)0, c, false, false);
        pS += 4; pB += 4;
    }
    return c;
}

// ---------------- layers 0/1: GEMM + bias + ReLU + L2-normalize, fully fused
// block = 256 threads = 8 waves = one M-tile (16 nodes) x all 128 cols.
__global__ void sage_gemm_fused_kernel(const float* __restrict__ hc,
                                       const float* __restrict__ s,
                                       const float* __restrict__ deg,
                                       const float* __restrict__ W,
                                       const float* __restrict__ bias,
                                       float* __restrict__ out) {
    __shared__ float part[8][16];   // per-wave partial sum-of-squares per row
    __shared__ float scale[16];     // final 1/norm per row

    int mTile  = blockIdx.x;
    int waveId = threadIdx.x >> 5;            // == nTile (8 waves x 16 cols)
    int lane   = threadIdx.x & 31;
    int half   = lane >> 4;
    int l16    = lane & 15;
    int arow   = mTile * 16 + l16;
    int brow   = waveId * 16 + l16;           // weight row == output column

    float invd = 1.0f / fmaxf(deg[arow], 1.0f);
    v8f c = wmma_tile_256(hc, s, invd, arow, brow, W, half);

    // bias + ReLU; deterministic block reduction of sum-of-squares per row.
    float bv = bias[brow];
    float vals[8];
    #pragma unroll
    for (int r = 0; r < 8; ++r) {
        float v = fmaxf(c[r] + bv, 0.0f);     // row = r + 8*half, col = brow
        vals[r] = v;
        float p = v * v;
        #pragma unroll
        for (int m = 8; m >= 1; m >>= 1)      // xor<16 stays within 16-group
            p += __shfl_xor(p, m, 32);
        if (l16 == 0) part[waveId][r + 8 * half] = p;   // lanes 0 and 16
    }
    __syncthreads();
    if (threadIdx.x < 16) {
        float t = 0.0f;
        #pragma unroll
        for (int wv = 0; wv < 8; ++wv) t += part[wv][threadIdx.x];  // fixed order
        scale[threadIdx.x] = 1.0f / fmaxf(sqrtf(t), 1e-12f);
    }
    __syncthreads();
    #pragma unroll
    for (int r = 0; r < 8; ++r) {
        int row16 = r + 8 * half;
        out[(size_t)(mTile * 16 + row16) * H_F + brow] = vals[r] * scale[row16];
    }
}

// ---------------- layer 2: plain GEMM + bias (ncols = 16), one wave per tile
__global__ void sage_gemm_out_kernel(const float* __restrict__ hc,
                                     const float* __restrict__ s,
                                     const float* __restrict__ deg,
                                     const float* __restrict__ W,
                                     const float* __restrict__ bias,
                                     float* __restrict__ out, int ntiles) {
    int wid = (blockIdx.x * blockDim.x + threadIdx.x) >> 5;
    if (wid >= ntiles) return;                // wave-uniform; EXEC stays all-1s
    int lane = threadIdx.x & 31;
    int half = lane >> 4;
    int l16  = lane & 15;
    int arow = wid * 16 + l16;                // ntilesN == 1
    int brow = l16;

    float invd = 1.0f / fmaxf(deg[arow], 1.0f);
    v8f c = wmma_tile_256(hc, s, invd, arow, brow, W, half);

    float bv = bias[brow];
    #pragma unroll
    for (int r = 0; r < 8; ++r) {
        int row = wid * 16 + r + 8 * half;
        out[(size_t)row * N_CLS + brow] = c[r] + bv;
    }
}

// ---------------------------------------------------------------------------
extern "C" void kernel_launch(void* const* d_in, const int* in_sizes, int n_in,
                              void* d_out, int out_size, void* d_ws, size_t ws_size,
                              hipStream_t stream) {
    (void)in_sizes; (void)n_in; (void)out_size; (void)ws_size;
    const float* h   = (const float*)d_in[0];
    const float* w   = (const float*)d_in[1];
    const int*   src = (const int*)d_in[2];
    const int*   dst = (const int*)d_in[3];
    const float* Wl[3] = {(const float*)d_in[4], (const float*)d_in[6], (const float*)d_in[8]};
    const float* bl[3] = {(const float*)d_in[5], (const float*)d_in[7], (const float*)d_in[9]};

    float* ws  = (float*)d_ws;
    float* s   = ws;                                   // N*128
    float* deg = s + (size_t)N_NODES * IN_F;           // N (contiguous after s)
    float* h1  = deg + N_NODES;                        // N*128
    float* h2  = h1 + (size_t)N_NODES * H_F;           // N*128

    const float* hc[3] = {h, h1, h2};
    float*     hout[2] = {h1, h2};

    for (int layer = 0; layer < 3; ++layer) {
        // zero accumulators (s and deg contiguous -> one launch)
        sage_zero_kernel<<<2048, 256, 0, stream>>>(s, N_NODES * IN_F + N_NODES);
        // scatter-sum + degree via native global_atomic_add_f32
        sage_aggregate_kernel<<<N_EDGES / 8, 256, 0, stream>>>(hc[layer], w, src, dst, s, deg);
        if (layer < 2) {
            // fused concat-GEMM + bias + ReLU + L2norm: block = one M-tile
            sage_gemm_fused_kernel<<<N_NODES / 16, 256, 0, stream>>>(
                hc[layer], s, deg, Wl[layer], bl[layer], hout[layer]);
        } else {
            int ntiles = N_NODES / 16;                 // 2500 waves
            sage_gemm_out_kernel<<<(ntiles + 7) / 8, 256, 0, stream>>>(
                hc[layer], s, deg, Wl[layer], bl[layer], (float*)d_out, ntiles);
        }
    }
}